// RR_44401371906493
// MI455X (gfx1250) — compile-verified
//
#include <hip/hip_runtime.h>
#include <stdint.h>

// Round-robin allocation, batched. X: [B, 32, 4096] f32 -> pi: [B, 32, 4096] f32 in {0,1}.
// One workgroup per batch. CDNA5: GLOBAL_LOAD_ASYNC_TO_LDS_B128 with a 2-deep pipeline
// (triple-buffered LDS rows, s_wait_asynccnt <= 4 keeps the newest row's DMA in flight).
// One barrier per pick step: packed (value,index) uint64 argmax, register taken mask.

#define NAG   32
#define M     4096
#define TPB   256
#define IPT   (M / TPB)      // 16 items per thread (one 16-bit mask slice)
#define NWAVE (TPB / 32)     // 8 waves

__device__ __forceinline__ void async_copy_row(const float* __restrict__ grow,
                                               uint32_t lds_base, int tid) {
  // 4096 floats = 1024 x b128 transfers; 4 per thread, wave-coalesced.
#pragma unroll
  for (int j = 0; j < 4; ++j) {
    const int e = (j * TPB + tid) * 4;                       // float index, 16B granule
    const unsigned long long g = (unsigned long long)(grow + e);
    const uint32_t l = lds_base + (uint32_t)(e * 4);         // LDS byte address
    asm volatile("global_load_async_to_lds_b128 %0, %1, off"
                 :: "v"(l), "v"(g) : "memory");
  }
}

__device__ __forceinline__ void wait_async0() {
  asm volatile("s_wait_asynccnt 0x0" ::: "memory");
}
__device__ __forceinline__ void wait_async4() {
  // Asyncs retire in order per wave: <=4 outstanding => the older row has landed,
  // the newest row's 4 b128 transfers may remain in flight (full-step overlap).
  asm volatile("s_wait_asynccnt 0x4" ::: "memory");
}

// Order-preserving float -> uint32 (no NaNs in input).
__device__ __forceinline__ uint32_t fkey(float f) {
  uint32_t u = __float_as_uint(f);
  return (u & 0x80000000u) ? ~u : (u | 0x80000000u);
}

__global__ __launch_bounds__(TPB) void rr_kernel(const float* __restrict__ X,
                                                 float* __restrict__ out) {
  __shared__ float              rowbuf[3][M];        // 48 KB triple buffer
  __shared__ unsigned char      owner[M];            // item -> agent
  __shared__ unsigned long long slots[2][NWAVE];     // per-step-parity wave results

  const int b   = blockIdx.x;
  const int tid = threadIdx.x;
  const float* Xb = X + (size_t)b * NAG * M;

  const uint32_t lb0 = (uint32_t)(uintptr_t)(&rowbuf[0][0]);
  const uint32_t lb1 = (uint32_t)(uintptr_t)(&rowbuf[1][0]);
  const uint32_t lb2 = (uint32_t)(uintptr_t)(&rowbuf[2][0]);

  // Prologue: rows 0 and 1 in flight; require row 0 resident (<=4 outstanding).
  async_copy_row(Xb + 0 * M, lb0, tid);
  async_copy_row(Xb + 1 * M, lb1, tid);
  wait_async4();
  __syncthreads();

  const int i0 = tid * IPT;        // this thread's 16 consecutive items
  unsigned int bits = 0u;          // register-resident taken mask for the slice

  int cur  = 0;                    // t % 3
  int nxt2 = 2;                    // (t+2) % 3

  for (int t = 0; t < M; ++t) {
    const int a = t & (NAG - 1);

    // Kick off DMA of row t+2 (two steps ahead) into the retiring buffer.
    if (t + 2 < M) {
      const int an = (t + 2) & (NAG - 1);
      const uint32_t lb = (nxt2 == 0) ? lb0 : ((nxt2 == 1) ? lb1 : lb2);
      async_copy_row(Xb + (size_t)an * M, lb, tid);
    }

    // ---- phase 1: local + wave argmax from LDS row ----
    const float* rowp = (cur == 0) ? &rowbuf[0][0]
                      : (cur == 1) ? &rowbuf[1][0] : &rowbuf[2][0];
    const float4* rp = (const float4*)(rowp + i0);
    float4 v0 = rp[0], v1 = rp[1], v2 = rp[2], v3 = rp[3];
    float vals[IPT] = {v0.x, v0.y, v0.z, v0.w, v1.x, v1.y, v1.z, v1.w,
                       v2.x, v2.y, v2.z, v2.w, v3.x, v3.y, v3.z, v3.w};

    float best = -INFINITY;
    int   bidx = i0;
#pragma unroll
    for (int k = 0; k < IPT; ++k) {
      const bool avail = ((bits >> k) & 1u) == 0u;
      const float v = vals[k];
      if (avail && v > best) { best = v; bidx = i0 + k; }  // strict > => first max wins
    }

    // pack: high = ordered value key, low = ~idx (max => max value, tie -> min idx)
    unsigned long long c =
        ((unsigned long long)fkey(best) << 32) | (uint32_t)(~(uint32_t)bidx);

    // wave32 butterfly max
#pragma unroll
    for (int off = 16; off > 0; off >>= 1) {
      const unsigned long long oc = __shfl_xor(c, off, 32);
      if (oc > c) c = oc;
    }
    const int cb = t & 1;
    if ((tid & 31) == 0) slots[cb][tid >> 5] = c;

    // Row t+1 must be resident after the barrier; row t+2's DMA may stay in flight.
    if (t + 2 < M) wait_async4(); else wait_async0();
    __syncthreads();

    // ---- phase 2: every thread redundantly reduces the 8 wave results ----
    unsigned long long w = slots[cb][0];
#pragma unroll
    for (int s = 1; s < NWAVE; ++s) {
      const unsigned long long o = slots[cb][s];
      if (o > w) w = o;
    }
    const int bi = (int)(~(uint32_t)w);     // winning item
    if ((bi >> 4) == tid) {                 // owner thread of that 16-item slice
      owner[bi] = (unsigned char)a;
      bits |= 1u << (bi & 15);
    }

    cur  = (cur  == 2) ? 0 : cur  + 1;
    nxt2 = (nxt2 == 2) ? 0 : nxt2 + 1;
  }

  __syncthreads();

  // Emit pi[b][a][i] = (owner[i] == a), coalesced float4 stores.
  float* outb = out + (size_t)b * NAG * M;
  for (int a = 0; a < NAG; ++a) {
    float* oa = outb + (size_t)a * M;
#pragma unroll
    for (int r = 0; r < M / (TPB * 4); ++r) {   // 4 iterations of 1024 items
      const int i = (r * TPB + tid) * 4;
      float4 o;
      o.x = (owner[i + 0] == (unsigned char)a) ? 1.0f : 0.0f;
      o.y = (owner[i + 1] == (unsigned char)a) ? 1.0f : 0.0f;
      o.z = (owner[i + 2] == (unsigned char)a) ? 1.0f : 0.0f;
      o.w = (owner[i + 3] == (unsigned char)a) ? 1.0f : 0.0f;
      *(float4*)(oa + i) = o;
    }
  }
}

extern "C" void kernel_launch(void* const* d_in, const int* in_sizes, int n_in,
                              void* d_out, int out_size, void* d_ws, size_t ws_size,
                              hipStream_t stream) {
  (void)n_in; (void)out_size; (void)d_ws; (void)ws_size;
  const float* X = (const float*)d_in[0];
  float* out = (float*)d_out;
  const int batches = in_sizes[0] / (NAG * M);   // 128 for the reference shapes
  rr_kernel<<<batches, TPB, 0, stream>>>(X, out);
}